// DPLoss_25683904430204
// MI455X (gfx1250) — compile-verified
//
#include <hip/hip_runtime.h>
#include <math.h>

#define N   4096
#define DXK 512
#define DZK 64
#define ST  512          // threads in the sort/rank block

typedef float v2f __attribute__((ext_vector_type(2)));
typedef float v8f __attribute__((ext_vector_type(8)));

// ---------------------------------------------------------------------------
// Row squared norms: out[i] = sum_k X[i][k]^2
// ---------------------------------------------------------------------------
__global__ void rownorm_kernel(const float* __restrict__ X,
                               float* __restrict__ out, int kdim) {
    int i = blockIdx.x * blockDim.x + threadIdx.x;
    if (i >= N) return;
    const float* r = X + (size_t)i * kdim;
    float s = 0.0f;
    for (int k = 0; k < kdim; k += 4) {
        float4 v = *(const float4*)(r + k);
        s += v.x * v.x + v.y * v.y + v.z * v.z + v.w * v.w;
    }
    out[i] = s;
}

// ---------------------------------------------------------------------------
// Pairwise Euclidean distance via fp32 WMMA Gram tiles.
// One wave computes one 16x16 tile of D; 8 waves per block. x is L2-resident.
// ---------------------------------------------------------------------------
__global__ void dist_wmma_kernel(const float* __restrict__ X,
                                 const float* __restrict__ nrm,
                                 float* __restrict__ D, int kdim) {
    int lane = threadIdx.x & 31;
    int wave = threadIdx.x >> 5;
    int t  = blockIdx.x * 8 + wave;     // tile id over (N/16)^2 = 65536 tiles
    int tm = t >> 8;                    // 256 tiles per dim
    int tn = t & 255;

    int mr   = lane & 15;
    int koff = (lane >> 4) << 1;        // 0 or 2
    const float* arow = X + (size_t)(tm * 16 + mr) * kdim + koff;
    const float* brow = X + (size_t)(tn * 16 + mr) * kdim + koff;

    v8f c = {0.f, 0.f, 0.f, 0.f, 0.f, 0.f, 0.f, 0.f};
#pragma unroll 4
    for (int k = 0; k < kdim; k += 4) {
        v2f a = *(const v2f*)(arow + k);
        v2f b = *(const v2f*)(brow + k);
        c = __builtin_amdgcn_wmma_f32_16x16x4_f32(
                /*neg_a=*/false, a, /*neg_b=*/false, b,
                /*c_mod=*/(short)0, c, /*reuse_a=*/false, /*reuse_b=*/false);
    }

    // C/D layout: VGPR r, lanes 0-15 -> M=r, lanes 16-31 -> M=r+8; N = lane&15
    int col   = tn * 16 + (lane & 15);
    int mbase = tm * 16 + ((lane >> 4) << 3);
    float nc  = nrm[col];
#pragma unroll
    for (int r = 0; r < 8; ++r) {
        int row  = mbase + r;
        float sq = nrm[row] + nc - 2.0f * c[r];
        float d  = (sq > 0.0f) ? sqrtf(sq) : 0.0f;
        D[(size_t)row * N + col] = d;
    }
}

// ---------------------------------------------------------------------------
// In-LDS bitonic sort of N u64 keys, ascending. 78 stages, 2048 pairs/stage,
// 4 pairs per thread per stage. Caller must __syncthreads() before entry.
// ---------------------------------------------------------------------------
__device__ __forceinline__ void bitonic_sort_lds(unsigned long long* skey) {
    const int tid = threadIdx.x;
    for (int k = 2; k <= N; k <<= 1) {
        for (int j = k >> 1; j > 0; j >>= 1) {
#pragma unroll
            for (int q = 0; q < N / (ST * 2); ++q) {
                int idx = tid + q * ST;                    // pair index 0..2047
                int i   = ((idx & ~(j - 1)) << 1) | (idx & (j - 1)); // bit j of i == 0
                int ixj = i | j;
                bool up = ((i & k) == 0);
                unsigned long long a = skey[i];
                unsigned long long b = skey[ixj];
                if ((a > b) == up) { skey[i] = b; skey[ixj] = a; }
            }
            __syncthreads();
        }
    }
}

// ---------------------------------------------------------------------------
// Per-row ranks via two bitonic sorts (exact stable-argsort semantics:
// unique keys (dist_bits<<32)|index, dist>=0 => bits order-monotonic)
// + per-row pairdist partial sum. One 512-thread block per row.
// ---------------------------------------------------------------------------
__global__ __launch_bounds__(ST) void rank_sort_kernel(
        const float* __restrict__ Dxm, const float* __restrict__ Dzm,
        float* __restrict__ rowRank,  float* __restrict__ rowPd) {
    __shared__ unsigned long long skey[N];   // 32 KB
    __shared__ unsigned int      sbz[N];     // 16 KB (stashed z bits)
    __shared__ unsigned short    srank[N];   //  8 KB (rank_x)
    __shared__ float             sred[ST];   //  2 KB

    const int tid = threadIdx.x;
    const int i   = blockIdx.x;
    const float* rx = Dxm + (size_t)i * N;
    const float* rz = Dzm + (size_t)i * N;

    // ---- load both rows once; pairdist partial; pack x-keys, stash z bits
    float pd = 0.0f;
#pragma unroll
    for (int q = 0; q < 2; ++q) {
        int j = (q * ST + tid) * 4;
        float4 vx = *(const float4*)(rx + j);
        float4 vz = *(const float4*)(rz + j);
        float d0 = vz.x - vx.x, d1 = vz.y - vx.y;
        float d2 = vz.z - vx.z, d3 = vz.w - vx.w;
        pd += d0 * d0 + d1 * d1 + d2 * d2 + d3 * d3;
        skey[j + 0] = ((unsigned long long)__float_as_uint(vx.x) << 32) | (unsigned)(j + 0);
        skey[j + 1] = ((unsigned long long)__float_as_uint(vx.y) << 32) | (unsigned)(j + 1);
        skey[j + 2] = ((unsigned long long)__float_as_uint(vx.z) << 32) | (unsigned)(j + 2);
        skey[j + 3] = ((unsigned long long)__float_as_uint(vx.w) << 32) | (unsigned)(j + 3);
        sbz[j + 0] = __float_as_uint(vz.x);
        sbz[j + 1] = __float_as_uint(vz.y);
        sbz[j + 2] = __float_as_uint(vz.z);
        sbz[j + 3] = __float_as_uint(vz.w);
    }
    __syncthreads();

    // ---- sort x keys; rank_x[j] = sorted position
    bitonic_sort_lds(skey);
#pragma unroll
    for (int q = 0; q < N / ST; ++q) {
        int p = tid + q * ST;
        unsigned j = (unsigned)skey[p];      // low 32 bits = original index
        srank[j] = (unsigned short)p;
    }
    __syncthreads();                         // all skey reads done

    // ---- rebuild keys from stashed z bits; sort; accumulate |rank diff|
#pragma unroll
    for (int q = 0; q < N / ST; ++q) {
        int t = tid + q * ST;
        skey[t] = ((unsigned long long)sbz[t] << 32) | (unsigned)t;
    }
    __syncthreads();
    bitonic_sort_lds(skey);

    int rsum = 0;
#pragma unroll
    for (int q = 0; q < N / ST; ++q) {
        int p = tid + q * ST;
        unsigned j = (unsigned)skey[p];
        int d = (int)srank[j] - p;           // rank_x[j] - rank_z[j]
        rsum += (d < 0) ? -d : d;
    }

    // ---- deterministic block reductions
    sred[tid] = (float)rsum;                 // row sum <= 4096*4095 < 2^24: exact
    __syncthreads();
    for (int s = ST / 2; s > 0; s >>= 1) {
        if (tid < s) sred[tid] += sred[tid + s];
        __syncthreads();
    }
    if (tid == 0) rowRank[i] = sred[0];
    __syncthreads();

    sred[tid] = pd;
    __syncthreads();
    for (int s = ST / 2; s > 0; s >>= 1) {
        if (tid < s) sred[tid] += sred[tid + s];
        __syncthreads();
    }
    if (tid == 0) rowPd[i] = sred[0];
}

// ---------------------------------------------------------------------------
// Final reduction over 4096 row partials (double accumulation), write losses.
// ---------------------------------------------------------------------------
__global__ void finalize_kernel(const float* __restrict__ rowRank,
                                const float* __restrict__ rowPd,
                                float* __restrict__ out) {
    __shared__ double sr[256];
    __shared__ double sp[256];
    int tid = threadIdx.x;
    double a = 0.0, b = 0.0;
#pragma unroll
    for (int q = 0; q < 16; ++q) {
        int j = tid + 256 * q;
        a += (double)rowRank[j];
        b += (double)rowPd[j];
    }
    sr[tid] = a;
    sp[tid] = b;
    __syncthreads();
    for (int s = 128; s > 0; s >>= 1) {
        if (tid < s) { sr[tid] += sr[tid + s]; sp[tid] += sp[tid + s]; }
        __syncthreads();
    }
    if (tid == 0) {
        const double invN2 = 1.0 / ((double)N * (double)N);
        double rank_loss = sr[0] * invN2 / 32.0;         // / K
        double pairdist  = sp[0] * invN2;
        out[0] = (float)(rank_loss + 0.5 * pairdist);    // LAMBDA_RANK=1, LAMBDA_PAIRDIST=0.5
        out[1] = (float)rank_loss;
        out[2] = (float)pairdist;
    }
}

// ---------------------------------------------------------------------------
extern "C" void kernel_launch(void* const* d_in, const int* in_sizes, int n_in,
                              void* d_out, int out_size, void* d_ws, size_t ws_size,
                              hipStream_t stream) {
    const float* x = (const float*)d_in[0];   // [4096, 512]
    const float* z = (const float*)d_in[1];   // [4096, 64]
    float* ws = (float*)d_ws;

    // Workspace layout (floats): 2*N*N dist matrices + norms + row partials.
    float* Dxm     = ws;                              // N*N
    float* Dzm     = Dxm + (size_t)N * N;             // N*N
    float* nx      = Dzm + (size_t)N * N;             // N
    float* nz      = nx + N;                          // N
    float* rowRank = nz + N;                          // N
    float* rowPd   = rowRank + N;                     // N

    rownorm_kernel<<<N / 256, 256, 0, stream>>>(x, nx, DXK);
    rownorm_kernel<<<N / 256, 256, 0, stream>>>(z, nz, DZK);

    // 65536 tiles of 16x16, 8 waves (tiles) per 256-thread block.
    dist_wmma_kernel<<<(256 * 256) / 8, 256, 0, stream>>>(x, nx, Dxm, DXK);
    dist_wmma_kernel<<<(256 * 256) / 8, 256, 0, stream>>>(z, nz, Dzm, DZK);

    rank_sort_kernel<<<N, ST, 0, stream>>>(Dxm, Dzm, rowRank, rowPd);
    finalize_kernel<<<1, 256, 0, stream>>>(rowRank, rowPd, (float*)d_out);
}